// TopologyNet_49263274885628
// MI455X (gfx1250) — compile-verified
//
#include <hip/hip_runtime.h>
#include <hip/hip_bf16.h>
#include <math.h>

// ---------------------------------------------------------------------------
// TopologyNet (DGCNN-like) for MI455X / gfx1250, wave32.
// Gram matrices and edge-conv GEMMs run on V_WMMA_F32_16X16X4_F32.
// LDS staging of center features uses GLOBAL_LOAD_ASYNC_TO_LDS_B32 (ASYNCcnt).
// ---------------------------------------------------------------------------

typedef __attribute__((ext_vector_type(2))) float v2f;
typedef __attribute__((ext_vector_type(8))) float v8f;

#define NPTS 2048
#define NB   8
#define KNN  20
#define SLOPE 0.2f
#define NEG_BIG (-3.0e38f)

__device__ __forceinline__ v8f wmma_f32_16x16x4(v2f a, v2f b, v8f c) {
  // 8 args: (neg_a, A, neg_b, B, c_mod, C, reuse_a, reuse_b)
  return __builtin_amdgcn_wmma_f32_16x16x4_f32(
      false, a, false, b, (short)0, c, false, false);
}

__device__ __forceinline__ int imin(int a, int b) { return a < b ? a : b; }

// Async global->LDS copy (4B per lane).  Generic LDS pointers hold the
// wave-relative LDS byte offset in their low 32 bits,
// which is exactly the VDST operand the async op wants.
__device__ __forceinline__ void async_ld_f32(float* ldsDst, const float* gsrc) {
  unsigned off = (unsigned)(unsigned long long)(void*)ldsDst;
  asm volatile("global_load_async_to_lds_b32 %0, %1, off"
               :: "v"(off), "v"((unsigned long long)gsrc)
               : "memory");
}
__device__ __forceinline__ void async_wait0() {
  asm volatile("s_wait_asynccnt 0" ::: "memory");
}

// ---------------------------------------------------------------------------
// x (B,N,3) row-major  ->  x0 (B,3,N)
// ---------------------------------------------------------------------------
__global__ void transpose_kernel(const float* __restrict__ xin,
                                 float* __restrict__ x0) {
  int b = blockIdx.y;
  int n = blockIdx.x * blockDim.x + threadIdx.x;
  if (n < NPTS) {
    for (int c = 0; c < 3; ++c)
      x0[((size_t)b * 3 + c) * NPTS + n] = xin[((size_t)b * NPTS + n) * 3 + c];
  }
}

// ---------------------------------------------------------------------------
// per-point squared norms: sq[b][n] = sum_c X[b][c][n]^2
// ---------------------------------------------------------------------------
template <int C>
__global__ void sqnorm_kernel(const float* __restrict__ X,
                              float* __restrict__ sq) {
  int b = blockIdx.y;
  int n = blockIdx.x * blockDim.x + threadIdx.x;
  if (n < NPTS) {
    const float* Xb = X + (size_t)b * C * NPTS;
    float s = 0.f;
#pragma unroll 8
    for (int c = 0; c < C; ++c) { float t = Xb[(size_t)c * NPTS + n]; s += t * t; }
    sq[(size_t)b * NPTS + n] = s;
  }
}

__global__ void fill_kernel(float* __restrict__ p, int n, float v) {
  int i = blockIdx.x * blockDim.x + threadIdx.x;
  if (i < n) p[i] = v;
}

// ---------------------------------------------------------------------------
// kNN: one workgroup = 16 rows of one batch.  8 waves each WMMA a 16x16 tile
// of the Gram matrix per 128-column chunk (tile stays in LDS), then 16 row-
// owner threads (one per wave-half) maintain a running sorted top-20
// (neg_dist = 2*G - sq[n] - sq[m]).
// ---------------------------------------------------------------------------
template <int C>
__global__ void knn_kernel(const float* __restrict__ X,
                           const float* __restrict__ sq,
                           int* __restrict__ idxout) {
  const int b = blockIdx.y;
  const int row0 = blockIdx.x * 16;
  const float* Xb = X + (size_t)b * C * NPTS;
  const float* sqb = sq + (size_t)b * NPTS;

  __shared__ float tile[16][128];
  __shared__ float topv[16][KNN];
  __shared__ int   topi[16][KNN];

  const int tid = threadIdx.x;
  const int wv = tid >> 5;
  const int lane = tid & 31;
  const int mlane = lane & 15;
  const int khalf = (lane >> 4) * 2;   // K offset {0 or 2} for A/B fragments
  const int rbase = (lane >> 4) * 8;   // M offset for C/D fragment rows

  // row r maintained by thread 16*r  (spreads rows over all 8 waves)
  const bool rowOwner = ((tid & 15) == 0);
  const int  myRow = tid >> 4;
  if (rowOwner) {
    for (int j = 0; j < KNN; ++j) { topv[myRow][j] = NEG_BIG; topi[myRow][j] = 0; }
  }
  __syncthreads();

  constexpr int KSTEPS = (C + 3) >> 2;
  const float* Arow = Xb + row0 + mlane;

  for (int chunk = 0; chunk < NPTS / 128; ++chunk) {
    const int col0 = chunk * 128 + wv * 16;
    const float* Bcol = Xb + col0 + mlane;
    v8f acc = {};
#pragma unroll 8
    for (int ks = 0; ks < KSTEPS; ++ks) {
      const int c0 = ks * 4 + khalf;
      v2f a, bb;
      if constexpr ((C & 3) == 0) {
        a.x  = Arow[(size_t)c0 * NPTS];
        a.y  = Arow[(size_t)(c0 + 1) * NPTS];
        bb.x = Bcol[(size_t)c0 * NPTS];
        bb.y = Bcol[(size_t)(c0 + 1) * NPTS];
      } else {
        // unconditional loads from clamped rows + cndmask select (no branches)
        const int cx = imin(c0, C - 1), cy = imin(c0 + 1, C - 1);
        const float axv = Arow[(size_t)cx * NPTS];
        const float ayv = Arow[(size_t)cy * NPTS];
        const float bxv = Bcol[(size_t)cx * NPTS];
        const float byv = Bcol[(size_t)cy * NPTS];
        a.x  = (c0     < C) ? axv : 0.f;
        a.y  = (c0 + 1 < C) ? ayv : 0.f;
        bb.x = (c0     < C) ? bxv : 0.f;
        bb.y = (c0 + 1 < C) ? byv : 0.f;
      }
      acc = wmma_f32_16x16x4(a, bb, acc);
    }
#pragma unroll
    for (int r = 0; r < 8; ++r) {
      const int m = r + rbase;
      const float d = 2.f * acc[r] - sqb[row0 + m] - sqb[col0 + mlane];
      tile[m][wv * 16 + mlane] = d;
    }
    __syncthreads();
    if (rowOwner) {
      const int base = chunk * 128;
      for (int j = 0; j < 128; ++j) {
        const float v = tile[myRow][j];
        if (v > topv[myRow][KNN - 1]) {
          int p = KNN - 1;
#pragma clang loop unroll(disable)
          while (p > 0 && topv[myRow][p - 1] < v) {
            topv[myRow][p] = topv[myRow][p - 1];
            topi[myRow][p] = topi[myRow][p - 1];
            --p;
          }
          topv[myRow][p] = v;
          topi[myRow][p] = base + j;
        }
      }
    }
    __syncthreads();
  }
  for (int t = tid; t < 16 * KNN; t += blockDim.x) {
    const int r = t / KNN, j = t % KNN;
    idxout[((size_t)b * NPTS + row0 + r) * KNN + j] = topi[r][j];
  }
}

// ---------------------------------------------------------------------------
// Edge conv:  y[o][(n,k)] = W[o][:] . edge[(n,k)][:], edge = [nb-ctr, ctr]
// One workgroup = one batch x 8 points (160 (n,k) pairs -> 10 column tiles).
// PASSB==0: accumulate per-channel sum/sum^2 for batch-norm.
// PASSB==1: recompute, normalize, leaky-relu, max over K into xout.
// Pairs >= 160 produce garbage columns that only affect discarded outputs;
// their neighbor index is clamped so addresses stay in-bounds.
// ---------------------------------------------------------------------------
template <int Cin, int O, int PASSB>
__global__ void edge_kernel(const float* __restrict__ X,
                            const int* __restrict__ nidx,
                            const float* __restrict__ W,
                            const float* __restrict__ mean,
                            const float* __restrict__ rstd,
                            const float* __restrict__ gamma,
                            const float* __restrict__ beta,
                            float* __restrict__ gsum,
                            float* __restrict__ gsq,
                            float* __restrict__ xout) {
  constexpr int twoC = 2 * Cin;
  const int b = blockIdx.y;
  const int n0 = blockIdx.x * 8;
  const float* Xb = X + (size_t)b * Cin * NPTS;

  __shared__ int   lidx[8 * KNN];   // neighbor indices for our 8 points
  __shared__ float cctr[Cin * 8];   // center features X[c][n0+j]
  __shared__ float lsum[O], lsq[O];
  __shared__ float red[O * 8];      // max-over-K accumulator [o][j]

  const int tid = threadIdx.x;
  const int wv = tid >> 5;
  const int lane = tid & 31;
  const int mlane = lane & 15;
  const int khalf = (lane >> 4) * 2;
  const int rbase = (lane >> 4) * 8;

  for (int t = tid; t < 8 * KNN; t += 256)
    lidx[t] = nidx[((size_t)b * NPTS + n0 + t / KNN) * KNN + (t % KNN)];
  // center features: async DMA into LDS (ASYNCcnt-tracked)
  for (int t = tid; t < Cin * 8; t += 256)
    async_ld_f32(&cctr[t], &Xb[(size_t)(t >> 3) * NPTS + n0 + (t & 7)]);
  for (int t = tid; t < O; t += 256) { lsum[t] = 0.f; lsq[t] = 0.f; }
  for (int t = tid; t < O * 8; t += 256) red[t] = NEG_BIG;
  async_wait0();
  __syncthreads();

  constexpr int NTILES = (O / 16) * 10;
  constexpr int KSTEPS = (twoC + 3) >> 2;

  for (int tix = wv; tix < NTILES; tix += 8) {
    const int otile = tix / 10;
    const int pt    = tix % 10;
    // B-fragment lane -> pair (column) id; clamp invalid pairs to pair 0
    const int p = pt * 16 + mlane;
    const bool pv = (p < 8 * KNN);
    const int j  = pv ? p / KNN : 0;
    const int kk = pv ? p % KNN : 0;
    const int nbv = lidx[j * KNN + kk];
    const float* Wrow = W + (size_t)(otile * 16 + mlane) * twoC;
    const float* Xnb  = Xb + nbv;

    v8f acc = {};
#pragma unroll 8
    for (int ks = 0; ks < KSTEPS; ++ks) {
      const int c0 = ks * 4 + khalf;
      v2f a, bb;
      if constexpr ((twoC & 3) == 0) {
        a = *(const v2f*)(Wrow + c0);               // contiguous pair -> b64
        if (c0 < Cin) {                              // both lanes same region
          __builtin_prefetch(Xnb + (size_t)(c0 + 8) * NPTS, 0, 1);
          bb.x = Xnb[(size_t)c0 * NPTS]       - cctr[c0 * 8 + j];
          bb.y = Xnb[(size_t)(c0 + 1) * NPTS] - cctr[(c0 + 1) * 8 + j];
        } else {
          bb.x = cctr[(c0 - Cin) * 8 + j];
          bb.y = cctr[(c0 + 1 - Cin) * 8 + j];
        }
      } else {
        // clamped unconditional loads + selects (no exec branching)
        const v2f av = *(const v2f*)(Wrow + imin(c0, twoC - 2));
        a = (c0 < twoC) ? av : (v2f){0.f, 0.f};
        const int c1 = c0 + 1;
        const float gx = Xnb[(size_t)imin(c0, Cin - 1) * NPTS];
        const float gy = Xnb[(size_t)imin(c1, Cin - 1) * NPTS];
        const float dx = gx - cctr[imin(c0, Cin - 1) * 8 + j];
        const float dy = gy - cctr[imin(c1, Cin - 1) * 8 + j];
        const float sx = cctr[imin(c0 - Cin < 0 ? 0 : c0 - Cin, Cin - 1) * 8 + j];
        const float sy = cctr[imin(c1 - Cin < 0 ? 0 : c1 - Cin, Cin - 1) * 8 + j];
        bb.x = (c0 < Cin) ? dx : ((c0 < twoC) ? sx : 0.f);
        bb.y = (c1 < Cin) ? dy : ((c1 < twoC) ? sy : 0.f);
      }
      acc = wmma_f32_16x16x4(a, bb, acc);
    }
    // D fragment: lane column = pair p; rows o = otile*16 + r + rbase
    if (pv) {
#pragma unroll
      for (int r = 0; r < 8; ++r) {
        const int o = otile * 16 + r + rbase;
        const float y = acc[r];
        if constexpr (PASSB == 0) {
          atomicAdd(&lsum[o], y);
          atomicAdd(&lsq[o], y * y);
        } else {
          float yn = (y - mean[o]) * rstd[o] * gamma[o] + beta[o];
          yn = (yn >= 0.f) ? yn : SLOPE * yn;
          atomicMax(&red[o * 8 + j], yn);
        }
      }
    }
  }
  __syncthreads();

  if constexpr (PASSB == 0) {
    for (int t = tid; t < O; t += 256) {
      atomicAdd(&gsum[t], lsum[t]);
      atomicAdd(&gsq[t],  lsq[t]);
    }
  } else {
    for (int t = tid; t < O * 8; t += 256)
      xout[((size_t)b * O + t / 8) * NPTS + n0 + (t & 7)] = red[t];
  }
}

__global__ void stats_kernel(const float* __restrict__ gsum,
                             const float* __restrict__ gsq,
                             float* __restrict__ mean,
                             float* __restrict__ rstd,
                             int O, float inv_cnt) {
  int o = blockIdx.x * blockDim.x + threadIdx.x;
  if (o < O) {
    float m = gsum[o] * inv_cnt;
    float v = gsq[o] * inv_cnt - m * m;
    mean[o] = m;
    rstd[o] = rsqrtf(v + 1e-5f);
  }
}

// ---------------------------------------------------------------------------
// global max pool over N for the concatenated 448-channel feature
// ---------------------------------------------------------------------------
__global__ void maxpool_kernel(const float* __restrict__ x1,
                               const float* __restrict__ x2,
                               const float* __restrict__ x3,
                               float* __restrict__ pooled) {
  const int ch = blockIdx.x;
  const int b  = blockIdx.y;
  const float* src;
  if (ch < 64)        src = x1 + ((size_t)b * 64  + ch)        * NPTS;
  else if (ch < 192)  src = x2 + ((size_t)b * 128 + (ch - 64)) * NPTS;
  else                src = x3 + ((size_t)b * 256 + (ch - 192)) * NPTS;

  __shared__ float sm[256];
  float m = NEG_BIG;
  for (int i = threadIdx.x; i < NPTS; i += 256) m = fmaxf(m, src[i]);
  sm[threadIdx.x] = m;
  __syncthreads();
  for (int s = 128; s > 0; s >>= 1) {
    if (threadIdx.x < s) sm[threadIdx.x] = fmaxf(sm[threadIdx.x], sm[threadIdx.x + s]);
    __syncthreads();
  }
  if (threadIdx.x == 0) pooled[b * 448 + ch] = sm[0];
}

// ---------------------------------------------------------------------------
// tiny FC (M=8) — bandwidth-bound GEMV, plain VALU is appropriate here
// ---------------------------------------------------------------------------
__global__ void fc_kernel(const float* __restrict__ in,
                          const float* __restrict__ Wt,
                          float* __restrict__ out,
                          int B, int I, int O, int leaky) {
  int t = blockIdx.x * blockDim.x + threadIdx.x;
  if (t < B * O) {
    int b = t / O, o = t % O;
    const float* ib = in + (size_t)b * I;
    const float* wr = Wt + (size_t)o * I;
    float s = 0.f;
#pragma unroll 8
    for (int i = 0; i < I; ++i) s += ib[i] * wr[i];
    if (leaky) s = (s >= 0.f) ? s : SLOPE * s;
    out[t] = s;
  }
}

// ---------------------------------------------------------------------------
extern "C" void kernel_launch(void* const* d_in, const int* in_sizes, int n_in,
                              void* d_out, int out_size, void* d_ws, size_t ws_size,
                              hipStream_t stream) {
  const float* x  = (const float*)d_in[0];
  const float* W1 = (const float*)d_in[1];
  const float* g1 = (const float*)d_in[2];
  const float* b1 = (const float*)d_in[3];
  const float* W2 = (const float*)d_in[4];
  const float* g2 = (const float*)d_in[5];
  const float* b2 = (const float*)d_in[6];
  const float* W3 = (const float*)d_in[7];
  const float* g3 = (const float*)d_in[8];
  const float* b3 = (const float*)d_in[9];
  const float* L1 = (const float*)d_in[10];
  const float* L2 = (const float*)d_in[11];
  float* out = (float*)d_out;

  // workspace layout (floats)
  float* ws = (float*)d_ws;
  float* x0 = ws;                              // 8*3*2048
  float* x1 = x0 + (size_t)NB * 3   * NPTS;    // 8*64*2048
  float* x2 = x1 + (size_t)NB * 64  * NPTS;    // 8*128*2048
  float* x3 = x2 + (size_t)NB * 128 * NPTS;    // 8*256*2048
  float* sq = x3 + (size_t)NB * 256 * NPTS;    // 8*2048
  int*   idx = (int*)(sq + (size_t)NB * NPTS); // 8*2048*20 ints
  float* gsum = (float*)(idx + (size_t)NB * NPTS * KNN); // 256
  float* gsq  = gsum + 256;                    // 256
  float* mean = gsq + 256;                     // 256
  float* rstd = mean + 256;                    // 256
  float* pooled = rstd + 256;                  // 8*448
  float* hbuf   = pooled + NB * 448;           // 8*1024

  const float inv_cnt = 1.f / ((float)NB * (float)NPTS * (float)KNN);

  transpose_kernel<<<dim3(NPTS / 256, NB), 256, 0, stream>>>(x, x0);

  // ---- block 1: Cin=3, O=64 ----
  sqnorm_kernel<3><<<dim3(NPTS / 256, NB), 256, 0, stream>>>(x0, sq);
  knn_kernel<3><<<dim3(NPTS / 16, NB), 256, 0, stream>>>(x0, sq, idx);
  fill_kernel<<<2, 256, 0, stream>>>(gsum, 512, 0.f);
  edge_kernel<3, 64, 0><<<dim3(NPTS / 8, NB), 256, 0, stream>>>(
      x0, idx, W1, mean, rstd, g1, b1, gsum, gsq, x1);
  stats_kernel<<<1, 256, 0, stream>>>(gsum, gsq, mean, rstd, 64, inv_cnt);
  edge_kernel<3, 64, 1><<<dim3(NPTS / 8, NB), 256, 0, stream>>>(
      x0, idx, W1, mean, rstd, g1, b1, gsum, gsq, x1);

  // ---- block 2: Cin=64, O=128 ----
  sqnorm_kernel<64><<<dim3(NPTS / 256, NB), 256, 0, stream>>>(x1, sq);
  knn_kernel<64><<<dim3(NPTS / 16, NB), 256, 0, stream>>>(x1, sq, idx);
  fill_kernel<<<2, 256, 0, stream>>>(gsum, 512, 0.f);
  edge_kernel<64, 128, 0><<<dim3(NPTS / 8, NB), 256, 0, stream>>>(
      x1, idx, W2, mean, rstd, g2, b2, gsum, gsq, x2);
  stats_kernel<<<1, 256, 0, stream>>>(gsum, gsq, mean, rstd, 128, inv_cnt);
  edge_kernel<64, 128, 1><<<dim3(NPTS / 8, NB), 256, 0, stream>>>(
      x1, idx, W2, mean, rstd, g2, b2, gsum, gsq, x2);

  // ---- block 3: Cin=128, O=256 ----
  sqnorm_kernel<128><<<dim3(NPTS / 256, NB), 256, 0, stream>>>(x2, sq);
  knn_kernel<128><<<dim3(NPTS / 16, NB), 256, 0, stream>>>(x2, sq, idx);
  fill_kernel<<<2, 256, 0, stream>>>(gsum, 512, 0.f);
  edge_kernel<128, 256, 0><<<dim3(NPTS / 8, NB), 256, 0, stream>>>(
      x2, idx, W3, mean, rstd, g3, b3, gsum, gsq, x3);
  stats_kernel<<<1, 256, 0, stream>>>(gsum, gsq, mean, rstd, 256, inv_cnt);
  edge_kernel<128, 256, 1><<<dim3(NPTS / 8, NB), 256, 0, stream>>>(
      x2, idx, W3, mean, rstd, g3, b3, gsum, gsq, x3);

  maxpool_kernel<<<dim3(448, NB), 256, 0, stream>>>(x1, x2, x3, pooled);
  fc_kernel<<<(NB * 1024 + 255) / 256, 256, 0, stream>>>(pooled, L1, hbuf, NB, 448, 1024, 1);
  fc_kernel<<<(NB * 2500 + 255) / 256, 256, 0, stream>>>(hbuf, L2, out, NB, 1024, 2500, 0);
}